// AlignBlock_5609227289132
// MI455X (gfx1250) — compile-verified
//
#include <hip/hip_runtime.h>
#include <math.h>

// MI455X / gfx1250, wave32. All matrix math via V_WMMA_F32_16X16X4_F32 (fp32-exact).

typedef __attribute__((ext_vector_type(2))) float v2f;
typedef __attribute__((ext_vector_type(4))) float v4f;
typedef __attribute__((ext_vector_type(8))) float v8f;

namespace {
constexpr int kB  = 2;
constexpr int kC  = 16;    // CM == CR
constexpr int kH  = 16;
constexpr int kT  = 400;
constexpr int kF  = 161;
constexpr int kD  = 100;   // delay
constexpr int kFP = 176;   // F padded to 11 full 16-col tiles (pad cols zeroed)
constexpr int kTT = kT / 16;          // 25 t-tiles
constexpr int kFT = kFP / 16;         // 11 f-tiles
constexpr int kCFT = (kC * kF) / 16;  // 161 (c,f)-tiles

// workspace layout in floats
constexpr size_t kQKStride = (size_t)kB * kH * kT * kFP;  // one of Q or K
constexpr size_t kQOff = 0;
constexpr size_t kKOff = kQKStride;
constexpr size_t kVOff = 2 * kQKStride;
constexpr size_t kVSize = (size_t)kB * kH * kT * kD;
constexpr size_t kAOff = kVOff + kVSize;
}  // namespace

__device__ __forceinline__ v8f wmma_f32_k4(v2f a, v2f b, v8f c) {
  // (neg_a, A, neg_b, B, c_mod, C, reuse_a, reuse_b)
  return __builtin_amdgcn_wmma_f32_16x16x4_f32(false, a, false, b, (short)0, c,
                                               false, false);
}

__global__ void k_zero(float* __restrict__ p, size_t n) {
  size_t i = (size_t)blockIdx.x * blockDim.x + threadIdx.x;
  if (i < n) p[i] = 0.0f;
}

// Stage 1: Q[b,h,t,f] = W_mic x_mic + b_mic ; K likewise. One wave per
// (f-tile, t, which*b). 16(h) x 16(f) tile, K-loop over 16 channels in 4 steps.
__global__ void k_proj(const float* __restrict__ x_mic,
                       const float* __restrict__ x_ref,
                       const float* __restrict__ w_mic,
                       const float* __restrict__ b_mic,
                       const float* __restrict__ w_ref,
                       const float* __restrict__ b_ref,
                       float* __restrict__ qk) {
  const int lane = threadIdx.x;
  const int m = lane & 15;
  const int half = lane >> 4;
  const int f0 = blockIdx.x * 16;
  const int t = blockIdx.y;
  const int which = blockIdx.z >> 1;  // 0 = Q (mic), 1 = K (ref)
  const int b = blockIdx.z & 1;

  const float* x = which ? x_ref : x_mic;   // [B][C][T][F]
  const float* w = which ? w_ref : w_mic;   // [H][C]
  const float* bv = which ? b_ref : b_mic;  // [H]

  const int f = f0 + m;
  const bool fin = (f < kF);
  const int fc = fin ? f : (kF - 1);  // clamped, always-valid address

  // Unconditional bias vector loads (lane needs bv[half*8 + r], r=0..7).
  const v4f bias0 = *(const v4f*)(bv + half * 8);
  const v4f bias1 = *(const v4f*)(bv + half * 8 + 4);

  // per-lane base for B loads: x[b][c=0][t][fc]
  const float* xb = x + ((size_t)b * kC * kT + t) * kF + fc;
  constexpr size_t kCStride = (size_t)kT * kF;  // channel stride in x

  v8f acc = {0.f, 0.f, 0.f, 0.f, 0.f, 0.f, 0.f, 0.f};
#pragma unroll
  for (int c0 = 0; c0 < kC; c0 += 4) {
    const int ca = c0 + half * 2;
    v2f a;
    a.x = w[m * kC + ca];
    a.y = w[m * kC + ca + 1];
    // unconditional loads (clamped addr) + select -> v_cndmask, no EXEC branches
    const float lx = xb[(size_t)ca * kCStride];
    const float ly = xb[(size_t)(ca + 1) * kCStride];
    v2f bb;
    bb.x = fin ? lx : 0.f;
    bb.y = fin ? ly : 0.f;
    acc = wmma_f32_k4(a, bb, acc);
  }

  float* dst = qk + (which ? kKOff : kQOff) + (size_t)b * kH * kT * kFP +
               ((size_t)half * 8 * kT + t) * kFP + f;
#pragma unroll
  for (int r = 0; r < 8; ++r) {
    const float bias = (r < 4) ? bias0[r & 3] : bias1[r & 3];
    const float val = fin ? (acc[r] + bias) : 0.f;  // pad cols stay exactly 0
    dst[(size_t)r * kT * kFP] = val;
  }
}

// Stage 2: banded Gram. V[b,h,t,d] = sum_f Q[t,f]*K[s,f], s = t+d-99.
// One wave per (s-tile offset, t-tile, b*h); only 8 aligned s-tiles touch the band.
__global__ void k_gram(const float* __restrict__ qk, float* __restrict__ V) {
  const int ti = blockIdx.y;
  const int si = ti - 7 + (int)blockIdx.x;  // blockIdx.x in [0,8)
  if (si < 0) return;                       // uniform over the wave
  const int bh = blockIdx.z;                // b*kH + h
  const int lane = threadIdx.x;
  const int m = lane & 15;
  const int half = lane >> 4;

  const float* qrow = qk + kQOff + ((size_t)bh * kT + (ti * 16 + m)) * kFP;
  const float* krow = qk + kKOff + ((size_t)bh * kT + (si * 16 + m)) * kFP;

  v8f acc = {0.f, 0.f, 0.f, 0.f, 0.f, 0.f, 0.f, 0.f};
  for (int f0 = 0; f0 < kFP; f0 += 4) {
    const int fa = f0 + half * 2;
    const v2f a = *(const v2f*)(qrow + fa);  // A[t, f] = Q row
    const v2f b = *(const v2f*)(krow + fa);  // B[f, s] = K row (Gram)
    acc = wmma_f32_k4(a, b, acc);
  }

#pragma unroll
  for (int r = 0; r < 8; ++r) {
    const int t = ti * 16 + r + half * 8;
    const int s = si * 16 + m;
    const int d = s - t + (kD - 1);
    if (d >= 0 && d < kD) V[((size_t)bh * kT + t) * kD + d] = acc[r];
  }
}

// Stage 3: 5x3 conv over (t,d) with 16 input channels -> 1, then softmax over d.
// One 128-thread block per (t,b); thread = d.
__global__ void k_conv_softmax(const float* __restrict__ V,
                               const float* __restrict__ w_conv,
                               const float* __restrict__ b_conv,
                               float* __restrict__ A) {
  const int t = blockIdx.x;
  const int b = blockIdx.y;
  const int d = threadIdx.x;
  __shared__ float wsh[kH * 5 * 3];
  __shared__ float red[128];
  for (int i = threadIdx.x; i < kH * 5 * 3; i += blockDim.x) wsh[i] = w_conv[i];
  __syncthreads();

  float vc = -INFINITY;
  if (d < kD) {
    vc = b_conv[0];
    for (int h = 0; h < kH; ++h) {
      const float* vrow = V + (size_t)(b * kH + h) * kT * kD;
#pragma unroll
      for (int kt = 0; kt < 5; ++kt) {
        const int tt = t + kt - 4;
        if (tt < 0) continue;
#pragma unroll
        for (int kd = 0; kd < 3; ++kd) {
          const int dd = d + kd - 1;
          if (dd < 0 || dd >= kD) continue;
          vc += vrow[(size_t)tt * kD + dd] * wsh[(h * 5 + kt) * 3 + kd];
        }
      }
    }
  }
  red[threadIdx.x] = vc;
  __syncthreads();
  for (int s = 64; s > 0; s >>= 1) {
    if (threadIdx.x < s)
      red[threadIdx.x] = fmaxf(red[threadIdx.x], red[threadIdx.x + s]);
    __syncthreads();
  }
  const float mx = red[0];
  __syncthreads();
  const float e = (d < kD) ? __expf(vc - mx) : 0.f;
  red[threadIdx.x] = e;
  __syncthreads();
  for (int s = 64; s > 0; s >>= 1) {
    if (threadIdx.x < s) red[threadIdx.x] += red[threadIdx.x + s];
    __syncthreads();
  }
  const float inv = 1.0f / red[0];
  if (d < kD) A[((size_t)b * kT + t) * kD + d] = e * inv;
}

// Stage 4: out[b,c,t,f] = sum_s A_band[t,s] * x_ref[b,c,s,f], s in [t-99, t].
// Banded GEMM: M=t (16), N=(c,f) (16), K=s (8 aligned tiles x 16).
__global__ void k_out(const float* __restrict__ x_ref,
                      const float* __restrict__ A, float* __restrict__ out) {
  const int lane = threadIdx.x;
  const int m = lane & 15;
  const int half = lane >> 4;
  const int cf = blockIdx.x * 16 + m;  // 0..2575
  const int c = cf / kF;
  const int f = cf - c * kF;
  const int ti = blockIdx.y;
  const int b = blockIdx.z;
  const int trow = ti * 16 + m;

  const float* Arow = A + ((size_t)b * kT + trow) * kD;  // softmax row for t=trow
  const float* Xc = x_ref + ((size_t)b * kC + c) * kT * kF + f;

  v8f acc = {0.f, 0.f, 0.f, 0.f, 0.f, 0.f, 0.f, 0.f};
  const int si_lo = (ti >= 7) ? (ti - 7) : 0;
  for (int si = si_lo; si <= ti; ++si) {
#pragma unroll
    for (int k0 = 0; k0 < 16; k0 += 4) {
      const int sa = si * 16 + k0 + half * 2;
      const int d0 = sa - trow + (kD - 1);
      const int d1 = d0 + 1;
      // clamped unconditional loads + select (no EXEC branches)
      const int d0c = (d0 < 0) ? 0 : ((d0 > kD - 1) ? (kD - 1) : d0);
      const int d1c = (d1 < 0) ? 0 : ((d1 > kD - 1) ? (kD - 1) : d1);
      const float a0 = Arow[d0c];
      const float a1 = Arow[d1c];
      v2f a;
      a.x = (d0 >= 0 && d0 < kD) ? a0 : 0.f;
      a.y = (d1 >= 0 && d1 < kD) ? a1 : 0.f;
      v2f bb;
      bb.x = Xc[(size_t)sa * kF];
      bb.y = Xc[(size_t)(sa + 1) * kF];
      acc = wmma_f32_k4(a, bb, acc);
    }
  }

#pragma unroll
  for (int r = 0; r < 8; ++r) {
    const int t = ti * 16 + r + half * 8;
    out[(((size_t)b * kC + c) * kT + t) * kF + f] = acc[r];
  }
}

extern "C" void kernel_launch(void* const* d_in, const int* in_sizes, int n_in,
                              void* d_out, int out_size, void* d_ws,
                              size_t ws_size, hipStream_t stream) {
  (void)in_sizes; (void)n_in; (void)out_size; (void)ws_size;
  const float* x_mic = (const float*)d_in[0];
  const float* x_ref = (const float*)d_in[1];
  const float* w_mic = (const float*)d_in[2];
  const float* b_mic = (const float*)d_in[3];
  const float* w_ref = (const float*)d_in[4];
  const float* b_ref = (const float*)d_in[5];
  const float* w_conv = (const float*)d_in[6];
  const float* b_conv = (const float*)d_in[7];
  // d_in[8] = delay (==100, compiled in)

  float* ws = (float*)d_ws;
  float* V = ws + kVOff;
  float* A = ws + kAOff;
  float* out = (float*)d_out;

  {  // V must be zero where the band / K-padding contributes nothing
    const size_t n = kVSize;
    const int blocks = (int)((n + 255) / 256);
    k_zero<<<blocks, 256, 0, stream>>>(V, n);
  }
  k_proj<<<dim3(kFT, kT, 2 * kB), 32, 0, stream>>>(x_mic, x_ref, w_mic, b_mic,
                                                   w_ref, b_ref, ws);
  k_gram<<<dim3(8, kTT, kB * kH), 32, 0, stream>>>(ws, V);
  k_conv_softmax<<<dim3(kT, kB), 128, 0, stream>>>(V, w_conv, b_conv, A);
  k_out<<<dim3(kCFT, kTT, kB), 32, 0, stream>>>(x_ref, A, out);
}